// Involution2D_635655160371
// MI455X (gfx1250) — compile-verified
//
#include <hip/hip_runtime.h>
#include <hip/hip_bf16.h>

typedef __attribute__((ext_vector_type(2))) float v2f;
typedef __attribute__((ext_vector_type(8))) float v8f;

#define CC   256   // channels
#define HID  64    // hidden
#define GKK  144   // groups * k*k
#define HH   64
#define WW   64
#define HW   4096  // H*W
#define NB   8     // batch
#define NGRP 16

// ---------------------------------------------------------------------------
// CDNA5 async memory->LDS helpers (GLOBAL_LOAD_ASYNC_TO_LDS_*, ASYNCcnt).
// Inline asm per CDNA5 ISA — portable across toolchains.
// LDS byte address = low 32 bits of the generic pointer (ISA 10.2).
// ---------------------------------------------------------------------------
__device__ __forceinline__ void async_ld_b32(void* lds_dst, const void* gsrc) {
    unsigned lds_off = (unsigned)(uintptr_t)lds_dst;
    unsigned long long ga = (unsigned long long)(uintptr_t)gsrc;
    asm volatile("global_load_async_to_lds_b32 %0, %1, off"
                 :: "v"(lds_off), "v"(ga) : "memory");
}
__device__ __forceinline__ void wait_async0() {
    asm volatile("s_wait_asynccnt 0" ::: "memory");
}

// ---------------------------------------------------------------------------
// Kernel 1: fused kernel-generator.
//   hid  = relu(Wr[64,256] @ X[256,4096] + br)   (per batch)
//   Kmap = Ws[144,64] @ hid + bs                 -> workspace [B,144,HW]
// Workgroup = 256 threads = 8 waves; each wave owns a 16-pixel N-tile.
// hid lives only in LDS (32 KB per workgroup).
// ---------------------------------------------------------------------------
__global__ __launch_bounds__(256)
void involution_gen_kernel(const float* __restrict__ x,        // [B, C, HW]
                           const float* __restrict__ w_reduce, // [64, 256]
                           const float* __restrict__ b_reduce, // [64]
                           const float* __restrict__ w_span,   // [144, 64]
                           const float* __restrict__ b_span,   // [144]
                           float* __restrict__ kmap)           // [B, 144, HW]
{
    __shared__ float hid_lds[8][HID][16]; // 8 waves x 64 x 16 floats = 32 KB

    const int tid  = threadIdx.x;
    const int wave = tid >> 5;
    const int lane = tid & 31;
    const int l16  = lane & 15;
    const int hi   = lane >> 4;          // 0: K={0,1}, 1: K={2,3}

    const int wg    = blockIdx.x;        // B * 32 workgroups
    const int b     = wg >> 5;           // / (4096/128)
    const int ptile = wg & 31;
    const int n0    = ptile * 128 + wave * 16;   // base pixel of this wave's tile

    const float* xb = x + (size_t)b * CC * HW;

    // ---------------- Stage 1: hid = relu(Wr @ X + br) ----------------
    v8f acc0 = {}, acc1 = {}, acc2 = {}, acc3 = {};
    #pragma unroll 4
    for (int kt = 0; kt < CC / 4; ++kt) {
        const int kbase = kt * 4 + 2 * hi;
        // B operand: X[k, n]  (one 16-pixel column tile)
        v2f bv;
        bv.x = xb[(size_t)(kbase + 0) * HW + n0 + l16];
        bv.y = xb[(size_t)(kbase + 1) * HW + n0 + l16];
        // A operand: Wr rows m0+l16, K pair
        const float* wr = w_reduce + (size_t)l16 * CC + kbase;
        v2f a0, a1, a2, a3;
        a0.x = wr[0 * 16 * CC + 0]; a0.y = wr[0 * 16 * CC + 1];
        a1.x = wr[1 * 16 * CC + 0]; a1.y = wr[1 * 16 * CC + 1];
        a2.x = wr[2 * 16 * CC + 0]; a2.y = wr[2 * 16 * CC + 1];
        a3.x = wr[3 * 16 * CC + 0]; a3.y = wr[3 * 16 * CC + 1];
        acc0 = __builtin_amdgcn_wmma_f32_16x16x4_f32(false, a0, false, bv, (short)0, acc0, false, false);
        acc1 = __builtin_amdgcn_wmma_f32_16x16x4_f32(false, a1, false, bv, (short)0, acc1, false, false);
        acc2 = __builtin_amdgcn_wmma_f32_16x16x4_f32(false, a2, false, bv, (short)0, acc2, false, false);
        acc3 = __builtin_amdgcn_wmma_f32_16x16x4_f32(false, a3, false, bv, (short)0, acc3, false, false);
    }

    // bias + relu, park hid tile in LDS (C/D layout: row = r + 8*hi, col = l16)
    #pragma unroll
    for (int r = 0; r < 8; ++r) {
        int row0 = r + 8 * hi;
        float v0 = acc0[r] + b_reduce[ 0 + row0];
        float v1 = acc1[r] + b_reduce[16 + row0];
        float v2 = acc2[r] + b_reduce[32 + row0];
        float v3 = acc3[r] + b_reduce[48 + row0];
        hid_lds[wave][ 0 + row0][l16] = v0 > 0.f ? v0 : 0.f;
        hid_lds[wave][16 + row0][l16] = v1 > 0.f ? v1 : 0.f;
        hid_lds[wave][32 + row0][l16] = v2 > 0.f ? v2 : 0.f;
        hid_lds[wave][48 + row0][l16] = v3 > 0.f ? v3 : 0.f;
    }
    __syncthreads();   // cheap; guarantees cross-lane LDS visibility

    // ---------------- Stage 2: Kmap = Ws @ hid + bs ----------------
    float* kout = kmap + (size_t)b * GKK * HW;
    for (int mt = 0; mt < 9; ++mt) {
        v8f kacc = {};
        #pragma unroll 4
        for (int kt = 0; kt < HID / 4; ++kt) {
            const int kbase = kt * 4 + 2 * hi;
            v2f av, bv;
            const float* ws = w_span + (size_t)(mt * 16 + l16) * HID + kbase;
            av.x = ws[0];
            av.y = ws[1];
            bv.x = hid_lds[wave][kbase + 0][l16];
            bv.y = hid_lds[wave][kbase + 1][l16];
            kacc = __builtin_amdgcn_wmma_f32_16x16x4_f32(false, av, false, bv, (short)0, kacc, false, false);
        }
        #pragma unroll
        for (int r = 0; r < 8; ++r) {
            int row = mt * 16 + r + 8 * hi;
            kout[(size_t)row * HW + n0 + l16] = kacc[r] + b_span[row];
        }
    }
}

// ---------------------------------------------------------------------------
// Kernel 2: involution apply (3x3 location-specific stencil).
// One workgroup per (b, group, row). Kmap tile (9 x 64) and the group's
// 16-channel, 3-row padded x slab are staged into LDS with
// GLOBAL_LOAD_ASYNC_TO_LDS_B32 (no VGPR round-trip); halo stays zero from
// the pre-fill. Kmap is reused by all 16 channels of the group.
// ---------------------------------------------------------------------------
__global__ __launch_bounds__(256)
void involution_apply_kernel(const float* __restrict__ x,    // [B, C, H, W]
                             const float* __restrict__ kmap, // [B, 144, H, W]
                             float* __restrict__ out)        // [B, C, H, W]
{
    __shared__ float k_s[9][WW];          // 2.25 KB
    __shared__ float x_s[16][3][WW + 2];  // 12.4 KB

    const int blk = blockIdx.x;           // B * G * H = 8192
    const int h   = blk & 63;
    const int g   = (blk >> 6) & 15;
    const int b   = blk >> 10;

    const int tid = threadIdx.x;
    const int w   = tid & 63;
    const int cs  = tid >> 6;             // 0..3 channel slot

    // Phase 1: zero-fill the x slab (provides the stencil's zero halo).
    float* xflat = &x_s[0][0][0];
    for (int i = tid; i < 16 * 3 * (WW + 2); i += 256)
        xflat[i] = 0.f;
    __syncthreads();                      // zero-fill visible before async writes

    // Phase 2: async-stage Kmap[b, g*9 .. g*9+8, h, :]  (always in-bounds)
    const float* kb = kmap + ((size_t)b * GKK + g * 9) * HW + h * WW;
    for (int i = tid; i < 9 * WW; i += 256)
        async_ld_b32(&k_s[i / WW][i % WW], kb + (size_t)(i / WW) * HW + (i % WW));

    // async-stage x[b, g*16 .. g*16+15, h-1..h+1, :] (in-bounds lanes only;
    // EXEC predication skips halo lanes, which keep their zero fill)
    const float* xg = x + ((size_t)b * CC + g * 16) * HW;
    for (int i = tid; i < 16 * 3 * (WW + 2); i += 256) {
        int c   = i / (3 * (WW + 2));
        int rem = i % (3 * (WW + 2));
        int r   = rem / (WW + 2);         // 0..2 -> row h-1+r
        int cc  = rem % (WW + 2);         // 0..65 -> col cc-1
        int hh  = h - 1 + r;
        int ww  = cc - 1;
        if (hh >= 0 && hh < HH && ww >= 0 && ww < WW)
            async_ld_b32(&x_s[c][r][cc], xg + (size_t)c * HW + hh * WW + ww);
    }
    wait_async0();                        // drain this wave's ASYNCcnt
    __syncthreads();                      // all waves' async LDS writes visible

    float* ob = out + ((size_t)b * CC + g * 16) * HW + h * WW;
    #pragma unroll
    for (int it = 0; it < 4; ++it) {
        const int c = it * 4 + cs;        // channel within group
        float sum = 0.f;
        #pragma unroll
        for (int i = 0; i < 3; ++i)
            #pragma unroll
            for (int j = 0; j < 3; ++j)
                sum = fmaf(x_s[c][i][w + j], k_s[i * 3 + j][w], sum);
        ob[(size_t)c * HW + w] = sum;
    }
}

// ---------------------------------------------------------------------------
extern "C" void kernel_launch(void* const* d_in, const int* in_sizes, int n_in,
                              void* d_out, int out_size, void* d_ws, size_t ws_size,
                              hipStream_t stream)
{
    const float* x        = (const float*)d_in[0];
    const float* w_reduce = (const float*)d_in[1];
    const float* b_reduce = (const float*)d_in[2];
    const float* w_span   = (const float*)d_in[3];
    const float* b_span   = (const float*)d_in[4];
    float* out  = (float*)d_out;
    float* kmap = (float*)d_ws;   // B * 144 * 4096 floats = 18.9 MB scratch

    // Stage A: kernel generator (two fused WMMA GEMMs), 256 WGs x 256 threads
    involution_gen_kernel<<<NB * (HW / 128), 256, 0, stream>>>(
        x, w_reduce, b_reduce, w_span, b_span, kmap);

    // Stage B: apply involution, one WG per (b, group, row)
    involution_apply_kernel<<<NB * NGRP * HH, 256, 0, stream>>>(x, kmap, out);
}